// Xt_68796786147727
// MI455X (gfx1250) — compile-verified
//
#include <hip/hip_runtime.h>

typedef float f32x2 __attribute__((ext_vector_type(2)));
typedef float f32x4 __attribute__((ext_vector_type(4)));
typedef float f32x8 __attribute__((ext_vector_type(8)));

#define EPS   1e-5f
#define BATCH 4
#define CIN   256
#define K1    128      // P*H_DIM
#define PP    32
#define HD    4
#define WDIM  56
#define HW    3136     // 56*56
#define NHW   (BATCH*HW)  // 12544

__device__ __forceinline__ f32x8 wmma_f32_16x16x4(f32x2 a, f32x2 b, f32x8 c) {
  // D = A(16x4,f32) x B(4x16,f32) + C(16x16,f32); wave32, one tile per wave
  return __builtin_amdgcn_wmma_f32_16x16x4_f32(false, a, false, b, (short)0, c,
                                               false, false);
}

// ---------------------------------------------------------------------------
// Stage 1: y1 = w1[128x256] @ x[b][256x3136] + b1   (raw, pre-BN)
// One wave per 16(M) x 32(N) block: the A tile (w1) is reused across both
// N-tiles in registers -> 3 loads per WMMA instead of 4. K=256 = 64 K-steps.
// ---------------------------------------------------------------------------
__global__ void k_gemm1(const float* __restrict__ x, const float* __restrict__ w1,
                        const float* __restrict__ b1, float* __restrict__ y1) {
  const int lane = threadIdx.x;
  const bool hi  = lane >= 16;
  const int  l   = lane & 15;
  const int  nb  = blockIdx.x * 32;   // spatial (two 16-wide tiles)
  const int  mb  = blockIdx.y * 16;   // output-channel tile
  const int  bb  = blockIdx.z;

  f32x8 c0, c1;
#pragma unroll
  for (int r = 0; r < 8; ++r) {
    const float bias = b1[mb + r + (hi ? 8 : 0)];
    c0[r] = bias;
    c1[r] = bias;
  }

  const int arow  = mb + l;  // A: lanes 0-15 hold K0,K1; 16-31 hold K2,K3
  const int bcol0 = nb + l;  // B: mirrored layout
  const float* xb = x + (size_t)bb * CIN * HW;

  for (int k = 0; k < CIN; k += 4) {
    const int kk = k + (hi ? 2 : 0);
    f32x2 a, b0, b1v;
    a.x   = w1[arow * CIN + kk];
    a.y   = w1[arow * CIN + kk + 1];
    const float* r0 = xb + (size_t)kk * HW;
    const float* r1 = xb + (size_t)(kk + 1) * HW;
    b0.x  = r0[bcol0];       b0.y  = r1[bcol0];
    b1v.x = r0[bcol0 + 16];  b1v.y = r1[bcol0 + 16];
    c0 = wmma_f32_16x16x4(a, b0,  c0);
    c1 = wmma_f32_16x16x4(a, b1v, c1);
  }

  float* yb = y1 + (size_t)bb * K1 * HW;
#pragma unroll
  for (int r = 0; r < 8; ++r) {
    const int row = mb + r + (hi ? 8 : 0);
    yb[(size_t)row * HW + bcol0]      = c0[r];
    yb[(size_t)row * HW + bcol0 + 16] = c1[r];
  }
}

// ---------------------------------------------------------------------------
// Per-channel batch stats over (N,H,W) -> fold BN into (a,c): bn(v)=v*a+c
// src layout [B][nchan][HW]. One block per channel.
// ---------------------------------------------------------------------------
__global__ void k_stats(const float* __restrict__ src, int nchan,
                        const float* __restrict__ g, const float* __restrict__ be,
                        float* __restrict__ outA, float* __restrict__ outC) {
  __shared__ float r1[256];
  __shared__ float r2[256];
  const int k = blockIdx.x;
  const int t = threadIdx.x;
  float s = 0.f, s2 = 0.f;
  for (int idx = t; idx < NHW; idx += 256) {
    const int b = idx / HW, i = idx - b * HW;
    const float v = src[((size_t)b * nchan + k) * HW + i];
    s += v; s2 = fmaf(v, v, s2);
  }
  r1[t] = s; r2[t] = s2;
  __syncthreads();
  for (int off = 128; off > 0; off >>= 1) {
    if (t < off) { r1[t] += r1[t + off]; r2[t] += r2[t + off]; }
    __syncthreads();
  }
  if (t == 0) {
    const float mean = r1[0] * (1.f / NHW);
    const float var  = r2[0] * (1.f / NHW) - mean * mean;  // biased, as torch BN
    const float a = g[k] * rsqrtf(var + EPS);
    outA[k] = a;
    outC[k] = be[k] - mean * a;
  }
}

// ---------------------------------------------------------------------------
// Stage 2: grouped 3x3 conv (groups=32, 4->4), input = relu(bn1(y1)) on the
// fly, output raw y2 (post-bias, pre-BN2). K=36 per output => VALU kernel.
// ---------------------------------------------------------------------------
__global__ void k_conv3x3(const float* __restrict__ y1, const float* __restrict__ w2,
                          const float* __restrict__ b2, const float* __restrict__ a1,
                          const float* __restrict__ c1, float* __restrict__ y2) {
  const int idx = blockIdx.x * blockDim.x + threadIdx.x;  // [0, B*K1*HW)
  const int i = idx % HW;
  const int k = (idx / HW) % K1;
  const int b = idx / (HW * K1);
  const int h = i / WDIM, w = i - h * WDIM;
  const int grp = k >> 2;

  float acc = b2[k];
#pragma unroll
  for (int hp = 0; hp < HD; ++hp) {
    const int ch = grp * HD + hp;
    const float sa = a1[ch], sc = c1[ch];
    const float* src = y1 + ((size_t)b * K1 + ch) * HW;
    const float* wk  = w2 + ((size_t)k * HD + hp) * 9;
#pragma unroll
    for (int dy = 0; dy < 3; ++dy) {
      const int hh = h + dy - 1;
      if (hh < 0 || hh >= WDIM) continue;
#pragma unroll
      for (int dx = 0; dx < 3; ++dx) {
        const int ww = w + dx - 1;
        if (ww < 0 || ww >= WDIM) continue;
        float v = src[hh * WDIM + ww];
        v = fmaxf(fmaf(v, sa, sc), 0.f);
        acc = fmaf(v, wk[dy * 3 + dx], acc);
      }
    }
  }
  y2[idx] = acc;
}

// ---------------------------------------------------------------------------
// Per-path moments of u = relu(bn2(y2)) (4-vector): E[u] (4) and E[u u^T] (16).
// y3 is affine in u, so stage-3 BN stats follow analytically -> never
// materialize the [B,P,C,H,W] (411 MB) tensor.
// ---------------------------------------------------------------------------
__global__ void k_pathstats(const float* __restrict__ y2, const float* __restrict__ a2,
                            const float* __restrict__ c2, float* __restrict__ pstat) {
  __shared__ float red[256];
  const int p = blockIdx.x;
  const int t = threadIdx.x;
  float acc[20];
#pragma unroll
  for (int q = 0; q < 20; ++q) acc[q] = 0.f;

  for (int idx = t; idx < NHW; idx += 256) {
    const int b = idx / HW, i = idx - b * HW;
    float u[HD];
#pragma unroll
    for (int hp = 0; hp < HD; ++hp) {
      const int ch = p * HD + hp;
      const float v = y2[((size_t)b * K1 + ch) * HW + i];
      u[hp] = fmaxf(fmaf(v, a2[ch], c2[ch]), 0.f);
      acc[hp] += u[hp];
    }
#pragma unroll
    for (int a_ = 0; a_ < HD; ++a_)
#pragma unroll
      for (int b_ = 0; b_ < HD; ++b_)
        acc[4 + a_ * 4 + b_] = fmaf(u[a_], u[b_], acc[4 + a_ * 4 + b_]);
  }

  for (int q = 0; q < 20; ++q) {
    red[t] = acc[q];
    __syncthreads();
    for (int off = 128; off > 0; off >>= 1) {
      if (t < off) red[t] += red[t + off];
      __syncthreads();
    }
    if (t == 0) pstat[p * 20 + q] = red[0] * (1.f / NHW);
    __syncthreads();
  }
}

// ---------------------------------------------------------------------------
// Stage-3 BN coefficients: mean3 = w3.m + b3; var3 = w3^T (M - m m^T) w3.
// One block per path, one thread per output channel.
// ---------------------------------------------------------------------------
__global__ void k_coef3(const float* __restrict__ w3, const float* __restrict__ b3,
                        const float* __restrict__ g3, const float* __restrict__ be3,
                        const float* __restrict__ pstat, float* __restrict__ a3,
                        float* __restrict__ c3) {
  const int p = blockIdx.x;
  const int c = threadIdx.x;
  const float* ps = pstat + p * 20;
  float wv[HD], m[HD];
#pragma unroll
  for (int i = 0; i < HD; ++i) {
    wv[i] = w3[((size_t)p * CIN + c) * HD + i];
    m[i]  = ps[i];
  }
  float mean = b3[p * CIN + c];
#pragma unroll
  for (int i = 0; i < HD; ++i) mean = fmaf(wv[i], m[i], mean);
  float var = 0.f;
#pragma unroll
  for (int i = 0; i < HD; ++i)
#pragma unroll
    for (int j = 0; j < HD; ++j)
      var = fmaf(wv[i] * wv[j], ps[4 + i * 4 + j] - m[i] * m[j], var);
  const float a = g3[p * CIN + c] * rsqrtf(var + EPS);
  a3[p * CIN + c] = a;
  c3[p * CIN + c] = be3[p * CIN + c] - mean * a;
}

// ---------------------------------------------------------------------------
// Stage 3 fused: per 16x16 tile, 32 WMMAs (K=4 == H_DIM, one per path),
// epilogue per path = affine(BN3)+ReLU, accumulate path-sum, add shortcut x.
// Coefficient reads (b3/a3/c3) are 8 contiguous floats -> b128 vector loads.
// Writes pre-BN4 sum "s" directly into d_out.
// ---------------------------------------------------------------------------
__global__ void k_gemm3(const float* __restrict__ x, const float* __restrict__ y2,
                        const float* __restrict__ w3, const float* __restrict__ b3,
                        const float* __restrict__ a2, const float* __restrict__ c2,
                        const float* __restrict__ a3, const float* __restrict__ c3,
                        float* __restrict__ out) {
  const int lane = threadIdx.x;
  const bool hi  = lane >= 16;
  const int  l   = lane & 15;
  const int  nb  = blockIdx.x * 16;   // spatial tile
  const int  mb  = blockIdx.y * 16;   // C tile
  const int  bb  = blockIdx.z;

  float acc[8];
#pragma unroll
  for (int r = 0; r < 8; ++r) acc[r] = 0.f;

  const int arow = mb + l;
  const int bcol = nb + l;
  const int koff = hi ? 2 : 0;
  const int roff = mb + (hi ? 8 : 0);     // first of 8 contiguous coef rows
  const float* y2b = y2 + (size_t)bb * K1 * HW;

#pragma unroll 2
  for (int p = 0; p < PP; ++p) {
    f32x2 a, b;
    a.x = w3[((size_t)p * CIN + arow) * HD + koff];
    a.y = w3[((size_t)p * CIN + arow) * HD + koff + 1];
    const int ch0 = p * HD + koff;
    const float v0 = y2b[(size_t)ch0 * HW + bcol];
    const float v1 = y2b[(size_t)(ch0 + 1) * HW + bcol];
    b.x = fmaxf(fmaf(v0, a2[ch0], c2[ch0]), 0.f);
    b.y = fmaxf(fmaf(v1, a2[ch0 + 1], c2[ch0 + 1]), 0.f);

    // 8 contiguous coefficients per lane -> two 128-bit loads per array
    const int cbase = p * CIN + roff;   // 32B-aligned (multiple of 8 floats)
    const f32x4 bb0 = *(const f32x4*)(b3 + cbase);
    const f32x4 bb1 = *(const f32x4*)(b3 + cbase + 4);
    const f32x4 aa0 = *(const f32x4*)(a3 + cbase);
    const f32x4 aa1 = *(const f32x4*)(a3 + cbase + 4);
    const f32x4 cc0 = *(const f32x4*)(c3 + cbase);
    const f32x4 cc1 = *(const f32x4*)(c3 + cbase + 4);

    f32x8 cc;
#pragma unroll
    for (int r = 0; r < 4; ++r) { cc[r] = bb0[r]; cc[r + 4] = bb1[r]; }

    const f32x8 d = wmma_f32_16x16x4(a, b, cc);

#pragma unroll
    for (int r = 0; r < 4; ++r) {
      acc[r]     += fmaxf(fmaf(d[r],     aa0[r], cc0[r]), 0.f);
      acc[r + 4] += fmaxf(fmaf(d[r + 4], aa1[r], cc1[r]), 0.f);
    }
  }

  const float* xb = x + (size_t)bb * CIN * HW;
  float* ob = out + (size_t)bb * CIN * HW;
#pragma unroll
  for (int r = 0; r < 8; ++r) {
    const int row = mb + r + (hi ? 8 : 0);
    const size_t o = (size_t)row * HW + nb + l;
    ob[o] = acc[r] + xb[o];
  }
}

// ---------------------------------------------------------------------------
// Final BN applied in place on d_out.
// ---------------------------------------------------------------------------
__global__ void k_bn4(float* __restrict__ out, const float* __restrict__ a4,
                      const float* __restrict__ c4) {
  const int idx = blockIdx.x * blockDim.x + threadIdx.x;
  const int ch = (idx / HW) % CIN;
  out[idx] = fmaf(out[idx], a4[ch], c4[ch]);
}

extern "C" void kernel_launch(void* const* d_in, const int* in_sizes, int n_in,
                              void* d_out, int out_size, void* d_ws, size_t ws_size,
                              hipStream_t stream) {
  (void)in_sizes; (void)n_in; (void)out_size; (void)ws_size;
  const float* x   = (const float*)d_in[0];
  const float* w1  = (const float*)d_in[1];
  const float* b1  = (const float*)d_in[2];
  const float* g1  = (const float*)d_in[3];
  const float* be1 = (const float*)d_in[4];
  const float* w2  = (const float*)d_in[5];
  const float* b2  = (const float*)d_in[6];
  const float* g2  = (const float*)d_in[7];
  const float* be2 = (const float*)d_in[8];
  const float* w3  = (const float*)d_in[9];
  const float* b3  = (const float*)d_in[10];
  const float* g3  = (const float*)d_in[11];
  const float* be3 = (const float*)d_in[12];
  const float* g4  = (const float*)d_in[13];
  const float* be4 = (const float*)d_in[14];
  float* out = (float*)d_out;

  float* ws = (float*)d_ws;
  const size_t SZ1 = (size_t)BATCH * K1 * HW;  // 1,605,632 floats
  float* y1    = ws;
  float* y2    = ws + SZ1;
  float* a1    = ws + 2 * SZ1;
  float* c1    = a1 + K1;
  float* a2    = c1 + K1;
  float* c2    = a2 + K1;
  float* a4    = c2 + K1;
  float* c4    = a4 + CIN;
  float* pstat = c4 + CIN;        // 32*20
  float* a3    = pstat + PP * 20; // 8192
  float* c3    = a3 + PP * CIN;   // 8192  -> total ws ~12.9 MB

  // Stage 1 GEMM (WMMA f32 16x16x4, K=256, A-tile reused across 2 N-tiles)
  k_gemm1<<<dim3(HW / 32, K1 / 16, BATCH), 32, 0, stream>>>(x, w1, b1, y1);
  // BN1 stats
  k_stats<<<K1, 256, 0, stream>>>(y1, K1, g1, be1, a1, c1);
  // Stage 2 grouped 3x3 (fused bn1+relu on input)
  k_conv3x3<<<(BATCH * K1 * HW) / 256, 256, 0, stream>>>(y1, w2, b2, a1, c1, y2);
  // BN2 stats
  k_stats<<<K1, 256, 0, stream>>>(y2, K1, g2, be2, a2, c2);
  // Per-path first/second moments of relu(bn2(y2))
  k_pathstats<<<PP, 256, 0, stream>>>(y2, a2, c2, pstat);
  // Analytic BN3 coefficients
  k_coef3<<<PP, CIN, 0, stream>>>(w3, b3, g3, be3, pstat, a3, c3);
  // Stage 3 fused GEMM (WMMA, K=4 per path) + BN3 + ReLU + path-sum + shortcut
  k_gemm3<<<dim3(HW / 16, CIN / 16, BATCH), 32, 0, stream>>>(x, y2, w3, b3,
                                                             a2, c2, a3, c3, out);
  // BN4 stats + in-place apply
  k_stats<<<CIN, 256, 0, stream>>>(out, CIN, g4, be4, a4, c4);
  k_bn4<<<(BATCH * CIN * HW) / 256, 256, 0, stream>>>(out, a4, c4);
}